// MyBNN_46231027974235
// MI455X (gfx1250) — compile-verified
//
#include <hip/hip_runtime.h>
#include <hip/hip_bf16.h>
#include <math.h>

#define NL 5
#define MF 50
#define MIX 5
#define BATCH 131072
#define MU_C 1.2566370614359173e-06f   /* 4*pi*1e-7 */
#define LOG2PI 1.8378770664093453f
#define TWO_PI 6.283185307179586f

typedef float v2f __attribute__((ext_vector_type(2)));
typedef float v8f __attribute__((ext_vector_type(8)));

// Full 32-lane f32 sum via V_WMMA_F32_16X16X4_F32.
// A (16x4): lane L vgpr0 = A[L%16, L<16?0:2] = p, vgpr1 = 0.
// B = ones(4x16), C = 0  =>  D[m,n] = p_m + p_{m+16}.
// Lane n (0..15) holds rows 0..7 in d[0..7]; lanes 16..31 hold rows 8..15.
// Sum of 8 regs + one cross-half shuffle gives the full wave sum in every lane.
__device__ __forceinline__ float wave_sum32(float p) {
  v2f a; a[0] = p;    a[1] = 0.0f;
  v2f b; b[0] = 1.0f; b[1] = 1.0f;
  v8f c = {};
  v8f d = __builtin_amdgcn_wmma_f32_16x16x4_f32(false, a, false, b,
                                                (short)0, c, false, false);
  float s = ((d[0] + d[1]) + (d[2] + d[3])) + ((d[4] + d[5]) + (d[6] + d[7]));
  s += __shfl_xor(s, 16, 32);
  return s;
}

// Raw v_rcp_f32 (~1 ulp); args here are finite, well-scaled positives.
__device__ __forceinline__ float frcp(float x) {
  return __builtin_amdgcn_rcpf(x);
}

__device__ __forceinline__ float softplus_f(float x) {
  return fmaxf(x, 0.0f) + log1pf(__expf(-fabsf(x)));   // exp arg in (-inf,0]
}

__global__ void __launch_bounds__(256)
elbo_kernel(const float* __restrict__ x,
            const float* __restrict__ s,
            const float* __restrict__ u_cat,
            const float* __restrict__ u_trunc,
            float* __restrict__ block_part) {
  const int b = blockIdx.x * 256 + threadIdx.x;
  const float* xr   = x + (size_t)b * (NL + 2 * MF);
  const float* srow = s + (size_t)b * (MIX * (2 * NL + 1));
  __builtin_prefetch(xr, 0, 0);           // global_prefetch_b8 (row reused 100x)

  // ---- mixture parameters (all static indices -> registers only) ----
  float pro[MIX], loc[MIX][NL], sig[MIX][NL], isg[MIX][NL];
  float sumsq = 0.f;
#pragma unroll
  for (int m = 0; m < MIX; ++m) {
    float v = srow[m * (2 * NL + 1) + 2 * NL];
    pro[m] = v * v;
    sumsq += pro[m];
  }
  float invs = 1.0f / sumsq;
#pragma unroll
  for (int m = 0; m < MIX; ++m) pro[m] *= invs;
#pragma unroll
  for (int m = 0; m < MIX; ++m)
#pragma unroll
    for (int l = 0; l < NL; ++l) {
      loc[m][l] = 4.0f / (1.0f + __expf(-srow[m * (2 * NL + 1) + l]));
      float sg  = softplus_f(srow[m * (2 * NL + 1) + NL + l]);
      sig[m][l] = sg;
      isg[m][l] = 1.0f / sg;              // one-time precise divide
    }

  // ---- categorical pick ----
  float u = u_cat[b];
  float cum = 0.f; int comp = 0;
#pragma unroll
  for (int m = 0; m < MIX; ++m) { cum += pro[m]; comp += (u > cum) ? 1 : 0; }
  if (comp > MIX - 1) comp = MIX - 1;

  // ---- select chosen component with cndmask chain (no dynamic indexing) ----
  float loc_s[NL], sig_s[NL], isg_s[NL];
#pragma unroll
  for (int l = 0; l < NL; ++l) {
    loc_s[l] = loc[0][l]; sig_s[l] = sig[0][l]; isg_s[l] = isg[0][l];
  }
#pragma unroll
  for (int m = 1; m < MIX; ++m) {
    bool pick = (comp == m);
#pragma unroll
    for (int l = 0; l < NL; ++l) {
      loc_s[l] = pick ? loc[m][l] : loc_s[l];
      sig_s[l] = pick ? sig[m][l] : sig_s[l];
      isg_s[l] = pick ? isg[m][l] : isg_s[l];
    }
  }

  // ---- truncated-normal samples ----
  float smp[NL];
#pragma unroll
  for (int l = 0; l < NL; ++l) {
    float lo = loc_s[l], sg = sig_s[l], rg = isg_s[l];
    float aa = (0.1f - lo) * rg;
    float bb = (3.9f - lo) * rg;
    float Fa = normcdff(aa), Fb = normcdff(bb);
    float ut = u_trunc[(size_t)b * NL + l];
    float z  = normcdfinvf(Fa + ut * (Fb - Fa));
    float q  = lo + sg * z;
    smp[l] = fminf(fmaxf(q, 0.1f), 3.9f);
  }

  // ---- entropy term: lq = logsumexp_m(log pro_m + comp_lp_m) ----
  float clp[MIX];
#pragma unroll
  for (int m = 0; m < MIX; ++m) {
    float acc = 0.f;
#pragma unroll
    for (int l = 0; l < NL; ++l) {
      float lo = loc[m][l], sg = sig[m][l], rg = isg[m][l];
      float zc = (smp[l] - lo) * rg;
      float am = (0.0f - lo) * rg;
      float bm = (4.0f - lo) * rg;
      float Z  = normcdff(bm) - normcdff(am);
      acc += -0.5f * zc * zc - logf(sg) - logf(Z);   // Z may be tiny: precise
    }
    clp[m] = logf(pro[m]) + acc - (float)NL * 0.5f * LOG2PI;
  }
  float mx = clp[0];
#pragma unroll
  for (int m = 1; m < MIX; ++m) mx = fmaxf(mx, clp[m]);
  float se = 0.f;
#pragma unroll
  for (int m = 0; m < MIX; ++m) se += __expf(clp[m] - mx);   // arg <= 0
  float lq = mx + logf(se);

  // ---- forward model + likelihood ----
  // dj = c/sqrt(r), wj = c*sqrt(r), t = c * (2*thick/sqrt(r)); c = sqrt(w*MU/2)
  float sqr[NL], tco[NL];
#pragma unroll
  for (int l = 0; l < NL; ++l) {
    float r = exp10f(smp[l]);
    sqr[l] = sqrtf(r);
    tco[l] = 2.0f * xr[l] / sqr[l];       // one-time precise divide
  }
  float sum_ll = 0.f;
  // Structural 5x unroll: outer loop kept rolled (10 trips) so the hot body
  // stays ~4KB (I$-resident) and VGPR pressure bounded; inner 5 independent
  // recursion chains give the ILP needed to cover trans-op latency.
#pragma clang loop unroll(disable)
  for (int fb = 0; fb < MF; fb += 5) {
#pragma unroll
    for (int ff = 0; ff < 5; ++ff) {
      const int f = fb + ff;
      float w   = TWO_PI * exp10f(-2.0f + 5.0f * (float)f / 49.0f);
      float cc  = sqrtf(w * MU_C * 0.5f);
      float imw = frcp(MU_C * w);
      float ZR = cc * sqr[NL - 1];
      float ZC = ZR;
#pragma unroll
      for (int j = NL - 2; j >= 0; --j) {
        float wj = cc * sqr[j];
        float t  = cc * tco[j];           // t <= ~0.12: raw HW trans ops safe
        float e  = __expf(-t);
        float st = __sinf(t);
        float ct = __cosf(t);
        float ejR = e * ct;
        float ejC = -e * st;
        float i0  = frcp((wj + ZR) * (wj + ZR) + (wj + ZC) * (wj + ZC));
        float rjR = (2.f * wj * wj - ZR * ZR - ZC * ZC) * i0;
        float rjC = (2.f * wj * (ZR - ZC)) * i0;
        float reR = rjR * ejR - rjC * ejC;
        float reC = rjR * ejC + rjC * ejR;
        float i2  = frcp((1.f + reR) * (1.f + reR) + reC * reC);
        float auxR = (1.f - reR * reR - reC * reC) * i2;
        float auxC = (-2.f * reC) * i2;
        float nZR = wj * (auxR - auxC);
        float nZC = wj * (auxC + auxR);
        ZR = nZR; ZC = nZC;
      }
      float absZ2 = ZR * ZR + ZC * ZC;
      float aRes  = __log10f(absZ2 * imw);
      float phas  = atan2f(ZC, ZR);
      {
        float sc = fabsf(0.03f * aRes);
        float r  = (aRes - xr[NL + f]) * frcp(sc);
        sum_ll += -0.5f * r * r - __logf(sc);  // sc: normal positive
      }
      {
        float sc = fabsf(0.03f * phas);
        float r  = (phas - xr[NL + MF + f]) * frcp(sc);
        sum_ll += -0.5f * r * r - __logf(sc);
      }
    }
  }
  sum_ll += -(float)(2 * MF) * 0.5f * LOG2PI;   // hoisted constant term

  float p = lq - 0.01f * sum_ll;   // per-element contribution (pre /B)

  // ---- block reduction: WMMA wave sum, then LDS across 8 waves ----
  float wsum = wave_sum32(p);      // EXEC all-1s here: full grid, no divergence
  __shared__ float lds[8];
  int lane = threadIdx.x & 31;
  int wid  = threadIdx.x >> 5;
  if (lane == 0) lds[wid] = wsum;
  __syncthreads();
  if (threadIdx.x == 0) {
    float t = 0.f;
#pragma unroll
    for (int i = 0; i < 8; ++i) t += lds[i];
    block_part[blockIdx.x] = t;
  }
}

__global__ void __launch_bounds__(256)
finish_kernel(const float* __restrict__ block_part,
              float* __restrict__ out, int nblk) {
  float p = 0.f;
  for (int i = threadIdx.x; i < nblk; i += 256) p += block_part[i];
  float wsum = wave_sum32(p);
  __shared__ float lds[8];
  int lane = threadIdx.x & 31;
  int wid  = threadIdx.x >> 5;
  if (lane == 0) lds[wid] = wsum;
  __syncthreads();
  if (threadIdx.x == 0) {
    float t = 0.f;
#pragma unroll
    for (int i = 0; i < 8; ++i) t += lds[i];
    out[0] = t / (float)BATCH + logf(4.0f);   // - term_p, term_p = -log 4
  }
}

extern "C" void kernel_launch(void* const* d_in, const int* in_sizes, int n_in,
                              void* d_out, int out_size, void* d_ws, size_t ws_size,
                              hipStream_t stream) {
  const float* x       = (const float*)d_in[0];
  const float* s       = (const float*)d_in[1];
  const float* u_cat   = (const float*)d_in[2];
  const float* u_trunc = (const float*)d_in[3];
  float* ws = (float*)d_ws;
  const int nblk = BATCH / 256;   // 512
  elbo_kernel<<<nblk, 256, 0, stream>>>(x, s, u_cat, u_trunc, ws);
  finish_kernel<<<1, 256, 0, stream>>>(ws, (float*)d_out, nblk);
}